// STFPMLoss_33560874451191
// MI455X (gfx1250) — compile-verified
//
#include <hip/hip_runtime.h>

typedef float v2f __attribute__((ext_vector_type(2)));
typedef float v8f __attribute__((ext_vector_type(8)));

// ---------------------------------------------------------------------------
// State layout (unsigned words) per layer, in ws:
//  [0] rank_rem   [1] prefix/vk_bits  [2] cnt_eq  [3] tie
//  [4] next_min_bits  [5] unused  [6] count_ge  [7] sum_ge(f32 bits)
//  [8] sum_n2(f32 bits)
// ---------------------------------------------------------------------------

__global__ __launch_bounds__(256)
void k_init(unsigned* __restrict__ z, int nwords, float* __restrict__ out) {
  int i = blockIdx.x * blockDim.x + threadIdx.x;
  if (i == 0) out[0] = 0.0f;
  for (; i < nwords; i += gridDim.x * blockDim.x) z[i] = 0u;
}

// ---------------------------------------------------------------------------
// Channel-norm reduction via V_WMMA_F32_16X16X4_F32.
// Wave handles 16 positions; A = x^2 (16 pos x 4 chan), B = ones(4x16).
// After C/4 accumulating WMMAs, every column of D = sum_c x^2 per row.
// A layout (ISA 7.12.2, 32-bit A 16x4): lanes 0-15 -> K=0,1 ; lanes 16-31 -> K=2,3.
// D layout: VGPR r -> row r (lanes 0-15) / row r+8 (lanes 16-31), N = lane&15.
// ---------------------------------------------------------------------------
__global__ __launch_bounds__(256)
void k_norms(const float* __restrict__ t, const float* __restrict__ s,
             float* __restrict__ invt, float* __restrict__ invs,
             int C, int planeShift, int cShift) {
  const int lane  = threadIdx.x & 31;
  const int wave  = threadIdx.x >> 5;
  const int m     = lane & 15;
  const int khalf = lane >> 4;
  const int plane = 1 << planeShift;

  const int posBase = (blockIdx.x * 8 + wave) * 16;
  const int pos = posBase + m;
  const int b  = pos >> planeShift;
  const int pp = pos & (plane - 1);
  const int base = ((b << cShift) << planeShift) + pp;  // element idx of (b, c=0, pp)

  v2f ones; ones.x = 1.0f; ones.y = 1.0f;
  v8f acct = {}; v8f accs = {};

  for (int c0 = 0; c0 < C; c0 += 4) {
    const int c = c0 + 2 * khalf;
    const float t0 = t[base + c * plane];
    const float t1 = t[base + (c + 1) * plane];
    const float s0 = s[base + c * plane];
    const float s1 = s[base + (c + 1) * plane];
    v2f at; at.x = t0 * t0; at.y = t1 * t1;
    v2f as; as.x = s0 * s0; as.y = s1 * s1;
    acct = __builtin_amdgcn_wmma_f32_16x16x4_f32(false, at, false, ones,
                                                 (short)0, acct, false, false);
    accs = __builtin_amdgcn_wmma_f32_16x16x4_f32(false, as, false, ones,
                                                 (short)0, accs, false, false);
  }

  if (m == 0) {  // lanes 0 and 16 hold column N=0 of D
#pragma unroll
    for (int r = 0; r < 8; ++r) {
      const int row = r + 8 * khalf;
      invt[posBase + row] = 1.0f / fmaxf(sqrtf(acct[r]), 1e-12f);
      invs[posBase + row] = 1.0f / fmaxf(sqrtf(accs[r]), 1e-12f);
    }
  }
}

// ---------------------------------------------------------------------------
// sum(n^2) reduction (b128 loads).
// ---------------------------------------------------------------------------
__global__ __launch_bounds__(256)
void k_n2(const float4* __restrict__ n4, unsigned* __restrict__ state, int N4) {
  __shared__ float red[256];
  float ls = 0.0f;
  for (int i = blockIdx.x * blockDim.x + threadIdx.x; i < N4;
       i += gridDim.x * blockDim.x) {
    float4 v = n4[i];
    ls += v.x * v.x + v.y * v.y + v.z * v.z + v.w * v.w;
  }
  red[threadIdx.x] = ls;
  __syncthreads();
  for (int o = 128; o > 0; o >>= 1) {
    if (threadIdx.x < o) red[threadIdx.x] += red[threadIdx.x + o];
    __syncthreads();
  }
  if (threadIdx.x == 0) atomicAdd((float*)&state[8], red[0]);
}

// ---------------------------------------------------------------------------
// Vectorized d recompute helper: 4 consecutive elements share (b,c) and map to
// 4 consecutive positions (plane is a multiple of 4), so t/s/invt/invs all use
// b128 loads.  i4 indexes float4 elements; N4 = N/4.
//   pp4  = i4 & (plane/4 - 1)
//   b    = i4 >> (planeShift + cShift - 2)
//   pos4 = (b << (planeShift-2)) + pp4     (float4 index into inv-norm arrays)
// ---------------------------------------------------------------------------
__device__ __forceinline__ float4 d_vec4(const float4* __restrict__ t4,
                                         const float4* __restrict__ s4,
                                         const float4* __restrict__ it4,
                                         const float4* __restrict__ is4,
                                         int i4, int planeShift, int cShift) {
  const int planeMask4 = (1 << (planeShift - 2)) - 1;
  const int pp4 = i4 & planeMask4;
  const int b   = i4 >> (planeShift + cShift - 2);
  const int pos4 = (b << (planeShift - 2)) + pp4;
  const float4 tv = t4[i4];
  const float4 sv = s4[i4];
  const float4 it = it4[pos4];
  const float4 is = is4[pos4];
  float4 d;
  float e;
  e = tv.x * it.x - sv.x * is.x; d.x = e * e;
  e = tv.y * it.y - sv.y * is.y; d.y = e * e;
  e = tv.z * it.z - sv.z * is.z; d.z = e * e;
  e = tv.w * it.w - sv.w * is.w; d.w = e * e;
  return d;
}

// ---------------------------------------------------------------------------
// Histogram pass of d-bits (radix-select level). predShift>=32 -> no predicate.
// ---------------------------------------------------------------------------
__global__ __launch_bounds__(256)
void k_hist(const float4* __restrict__ t4, const float4* __restrict__ s4,
            const float4* __restrict__ it4, const float4* __restrict__ is4,
            unsigned* __restrict__ hist, const unsigned* __restrict__ state,
            int N4, int planeShift, int cShift, int predShift, int binShift,
            int nbins) {
  __shared__ unsigned lh[2048];
  for (int i = threadIdx.x; i < nbins; i += blockDim.x) lh[i] = 0u;
  __syncthreads();

  const unsigned prefix = (predShift < 32) ? state[1] : 0u;
  const unsigned binMask = (unsigned)nbins - 1u;

  for (int i = blockIdx.x * blockDim.x + threadIdx.x; i < N4;
       i += gridDim.x * blockDim.x) {
    const float4 d = d_vec4(t4, s4, it4, is4, i, planeShift, cShift);
    const float dv[4] = {d.x, d.y, d.z, d.w};
#pragma unroll
    for (int j = 0; j < 4; ++j) {
      const unsigned bits = __float_as_uint(dv[j]);
      const unsigned hi = (predShift < 32) ? (bits >> predShift) : prefix;
      if (hi == prefix) atomicAdd(&lh[(bits >> binShift) & binMask], 1u);
    }
  }
  __syncthreads();
  for (int i = threadIdx.x; i < nbins; i += blockDim.x) {
    const unsigned v = lh[i];
    if (v) atomicAdd(&hist[i], v);
  }
}

// ---------------------------------------------------------------------------
// Tiny serial scan over one histogram level (single thread; 2048 iters max).
// ---------------------------------------------------------------------------
__global__ void k_scan(const unsigned* __restrict__ hist,
                       unsigned* __restrict__ state, int nbins, int width,
                       int isFinal, unsigned kInit, int useKInit) {
  unsigned krem = useKInit ? kInit : state[0];
  unsigned cum = 0, bin = (unsigned)nbins - 1, cnt = 0;
  for (int i = 0; i < nbins; ++i) {
    const unsigned c = hist[i];
    if (cum + c > krem) { bin = (unsigned)i; cnt = c; break; }
    cum += c;
  }
  const unsigned within = krem - cum;
  state[0] = within;
  state[1] = (state[1] << width) | bin;
  if (isFinal) {
    const unsigned tie = (within + 1u < cnt) ? 1u : 0u;
    state[2] = cnt;
    state[3] = tie;
    state[4] = tie ? state[1] : 0xFFFFFFFFu;  // pre-seed next_min for min pass
  }
}

// ---------------------------------------------------------------------------
// Minimum element strictly greater than v_k (the rank k+1 partner).
// ---------------------------------------------------------------------------
__global__ __launch_bounds__(256)
void k_minnext(const float4* __restrict__ t4, const float4* __restrict__ s4,
               const float4* __restrict__ it4, const float4* __restrict__ is4,
               unsigned* __restrict__ state, int N4, int planeShift,
               int cShift) {
  if (state[3]) return;  // tie: rank k+1 value == v_k (uniform branch)
  __shared__ unsigned red[256];
  const unsigned vk = state[1];
  unsigned lmin = 0xFFFFFFFFu;
  for (int i = blockIdx.x * blockDim.x + threadIdx.x; i < N4;
       i += gridDim.x * blockDim.x) {
    const float4 d = d_vec4(t4, s4, it4, is4, i, planeShift, cShift);
    const float dv[4] = {d.x, d.y, d.z, d.w};
#pragma unroll
    for (int j = 0; j < 4; ++j) {
      const unsigned bits = __float_as_uint(dv[j]);
      if (bits > vk && bits < lmin) lmin = bits;
    }
  }
  red[threadIdx.x] = lmin;
  __syncthreads();
  for (int o = 128; o > 0; o >>= 1) {
    if (threadIdx.x < o)
      red[threadIdx.x] = min(red[threadIdx.x], red[threadIdx.x + o]);
    __syncthreads();
  }
  if (threadIdx.x == 0 && red[0] != 0xFFFFFFFFu) atomicMin(&state[4], red[0]);
}

// ---------------------------------------------------------------------------
// Final pass: dhard = v_k + frac*(v_next - v_k);  sum & count of d >= dhard.
// ---------------------------------------------------------------------------
__global__ __launch_bounds__(256)
void k_mask(const float4* __restrict__ t4, const float4* __restrict__ s4,
            const float4* __restrict__ it4, const float4* __restrict__ is4,
            unsigned* __restrict__ state, int N4, int planeShift, int cShift,
            float frac) {
  __shared__ float redf[256];
  __shared__ unsigned redu[256];
  const float vk = __uint_as_float(state[1]);
  const unsigned nb = state[4];
  const float vn = (nb == 0xFFFFFFFFu) ? vk : __uint_as_float(nb);
  const float dhard = vk + frac * (vn - vk);

  float ls = 0.0f; unsigned lc = 0u;
  for (int i = blockIdx.x * blockDim.x + threadIdx.x; i < N4;
       i += gridDim.x * blockDim.x) {
    const float4 d = d_vec4(t4, s4, it4, is4, i, planeShift, cShift);
    const float dv[4] = {d.x, d.y, d.z, d.w};
#pragma unroll
    for (int j = 0; j < 4; ++j) {
      if (dv[j] >= dhard) { ls += dv[j]; lc += 1u; }
    }
  }
  redf[threadIdx.x] = ls; redu[threadIdx.x] = lc;
  __syncthreads();
  for (int o = 128; o > 0; o >>= 1) {
    if (threadIdx.x < o) {
      redf[threadIdx.x] += redf[threadIdx.x + o];
      redu[threadIdx.x] += redu[threadIdx.x + o];
    }
    __syncthreads();
  }
  if (threadIdx.x == 0) {
    atomicAdd((float*)&state[7], redf[0]);
    atomicAdd(&state[6], redu[0]);
  }
}

__global__ void k_finalize(const unsigned* __restrict__ state,
                           float* __restrict__ out, float invN) {
  const float sum_ge = __uint_as_float(state[7]);
  const float sum_n2 = __uint_as_float(state[8]);
  const float hard_mean = sum_ge / (float)state[6];
  out[0] += hard_mean + sum_n2 * invN;
}

// ---------------------------------------------------------------------------
extern "C" void kernel_launch(void* const* d_in, const int* in_sizes, int n_in,
                              void* d_out, int out_size, void* d_ws,
                              size_t ws_size, hipStream_t stream) {
  const float* T[3] = {(const float*)d_in[0], (const float*)d_in[3], (const float*)d_in[6]};
  const float* S[3] = {(const float*)d_in[1], (const float*)d_in[4], (const float*)d_in[7]};
  const float* Nt[3] = {(const float*)d_in[2], (const float*)d_in[5], (const float*)d_in[8]};
  float* out = (float*)d_out;
  char* ws = (char*)d_ws;

  const int Cs[3]  = {64, 128, 256};
  const int cSh[3] = {6, 7, 8};
  const int pSh[3] = {14, 12, 10};       // plane = H*W
  const int Ps[3]  = {16 << 14, 16 << 12, 16 << 10};  // B*H*W positions
  const int Ns[3]  = {16777216, 8388608, 4194304};

  // ws layout: [state+hists x3 (zeroed)] then [invnorm buffers] (~2.9 MB total)
  size_t off = 0;
  size_t stOff[3], h0Off[3], h1Off[3], h2Off[3];
  for (int l = 0; l < 3; ++l) {
    stOff[l] = off; off += 256;
    h0Off[l] = off; off += 2048 * 4;
    h1Off[l] = off; off += 2048 * 4;
    h2Off[l] = off; off += 2048 * 4;
  }
  const int zeroWords = (int)(off / 4);
  size_t itOff[3], isOff[3];
  for (int l = 0; l < 3; ++l) {
    itOff[l] = off; off += (size_t)Ps[l] * 4;
    isOff[l] = off; off += (size_t)Ps[l] * 4;
  }
  (void)ws_size; (void)in_sizes; (void)n_in; (void)out_size;

  k_init<<<64, 256, 0, stream>>>((unsigned*)ws, zeroWords, out);

  const int g = 4096;  // grid-stride element kernels
  for (int l = 0; l < 3; ++l) {
    unsigned* st = (unsigned*)(ws + stOff[l]);
    unsigned* h0 = (unsigned*)(ws + h0Off[l]);
    unsigned* h1 = (unsigned*)(ws + h1Off[l]);
    unsigned* h2 = (unsigned*)(ws + h2Off[l]);
    const float4* invt = (const float4*)(ws + itOff[l]);
    const float4* invs = (const float4*)(ws + isOff[l]);
    const int C = Cs[l], ps = pSh[l], cs = cSh[l], P = Ps[l], N = Ns[l];
    const int N4 = N / 4;
    const float4* t4 = (const float4*)T[l];
    const float4* s4 = (const float4*)S[l];

    const double qpos = 0.999 * (double)(N - 1);
    const unsigned k = (unsigned)qpos;
    const float frac = (float)(qpos - (double)k);

    k_norms<<<P / 128, 256, 0, stream>>>(T[l], S[l],
                                         (float*)(ws + itOff[l]),
                                         (float*)(ws + isOff[l]), C, ps, cs);
    k_n2<<<2048, 256, 0, stream>>>((const float4*)Nt[l], st, N / 4);

    // 3-level radix select on d bits: 11 | 11 | 10
    k_hist<<<g, 256, 0, stream>>>(t4, s4, invt, invs, h0, st, N4, ps, cs, 32, 21, 2048);
    k_scan<<<1, 1, 0, stream>>>(h0, st, 2048, 11, 0, k, 1);
    k_hist<<<g, 256, 0, stream>>>(t4, s4, invt, invs, h1, st, N4, ps, cs, 21, 10, 2048);
    k_scan<<<1, 1, 0, stream>>>(h1, st, 2048, 11, 0, 0u, 0);
    k_hist<<<g, 256, 0, stream>>>(t4, s4, invt, invs, h2, st, N4, ps, cs, 10, 0, 1024);
    k_scan<<<1, 1, 0, stream>>>(h2, st, 1024, 10, 1, 0u, 0);

    k_minnext<<<g, 256, 0, stream>>>(t4, s4, invt, invs, st, N4, ps, cs);
    k_mask<<<g, 256, 0, stream>>>(t4, s4, invt, invs, st, N4, ps, cs, frac);
    k_finalize<<<1, 1, 0, stream>>>(st, out, 1.0f / (float)N);
  }
}